// BilinearInterpolation_77292231459545
// MI455X (gfx1250) — compile-verified
//
#include <hip/hip_runtime.h>
#include <cstdint>

// Bilinear affine grid-sample, bandwidth-bound.
// X: (16, 512, 512, 16) f32, scale: (16,1), translate: (16,2) -> out (16,512,512,16) f32.
//
// Strategy: per 16x32 output tile the needed input region is a contiguous
// rectangle (axis-aligned affine, s in [0,1)). Stage it into LDS with the
// CDNA5 async-DMA path (global_load_async_to_lds_b128 / ASYNCcnt), sample
// the 2x2 neighborhoods from LDS (ds_load_b128), and stream results out with
// non-temporal 128-bit stores.

typedef float v4f __attribute__((ext_vector_type(4)));

#define TILE_H 16
#define TILE_W 32
#define RECT_H 20   // >= 16*max_s*512/511 + 2  (s < 1)
#define RECT_W 36   // >= 32*max_s*512/511 + 2
#define NTHREADS 256

__global__ __launch_bounds__(NTHREADS)
void bilinear_tile_kernel(const float* __restrict__ X,
                          const float* __restrict__ scale,
                          const float* __restrict__ translate,
                          float* __restrict__ out)
{
    __shared__ float smem[RECT_H * RECT_W * 16];   // 46,080 B

    const int b   = blockIdx.z;
    const int ty  = blockIdx.y;
    const int tx  = blockIdx.x;
    const int tid = threadIdx.x;

    // Per-batch affine params (uniform -> scalar loads).
    const float s  = scale[b];
    const float t0 = translate[2 * b + 0];
    const float t1 = translate[2 * b + 1];

    const int oy0 = ty * TILE_H;
    const int ox0 = tx * TILE_W;

    // grid coord c = -1 + i*(2/511); input coord = 0.5*(s*c + t + 1)*512
    auto mapx = [&](int ox) -> float {
        float xc = -1.0f + (float)ox * (2.0f / 511.0f);
        return 0.5f * (s * xc + t0 + 1.0f) * 512.0f;
    };
    auto mapy = [&](int oy) -> float {
        float yc = -1.0f + (float)oy * (2.0f / 511.0f);
        return 0.5f * (s * yc + t1 + 1.0f) * 512.0f;
    };

    // Monotone map (s >= 0): tile footprint is [xlo..xhi] x [ylo..yhi] after clipping.
    const int xlo = min(max((int)mapx(ox0), 0), 511);
    const int xhi = min(max((int)mapx(ox0 + TILE_W - 1) + 1, 0), 511);
    const int ylo = min(max((int)mapy(oy0), 0), 511);
    const int yhi = min(max((int)mapy(oy0 + TILE_H - 1) + 1, 0), 511);
    const int nCols = min(xhi - xlo + 1, RECT_W);
    const int nRows = min(yhi - ylo + 1, RECT_H);

    // ---- Stage input rectangle into LDS via CDNA5 async DMA path ----
    // Each 16B chunk: one global_load_async_to_lds_b128 (tracked by ASYNCcnt).
    const int chunksPerRow = nCols * 4;              // 4 x 16B per pixel (64 B)
    const int total = nRows * chunksPerRow;
    const uint64_t xbase = (uint64_t)(uintptr_t)X +
                           (uint64_t)b * (512ull * 512ull * 64ull);
    const uint32_t ldsBase = (uint32_t)(uintptr_t)(&smem[0]);   // low 32 bits = LDS offset

    for (int t = tid; t < total; t += NTHREADS) {
        int r  = t / chunksPerRow;
        int cr = t - r * chunksPerRow;
        uint64_t g = xbase +
                     (((uint64_t)(ylo + r) * 512ull + (uint64_t)xlo) * 64ull) +
                     (uint64_t)cr * 16ull;
        uint32_t l = ldsBase + (uint32_t)(r * (RECT_W * 64) + cr * 16);
        asm volatile("global_load_async_to_lds_b128 %0, %1, off"
                     :: "v"(l), "v"(g) : "memory");
    }
    // Wait for this wave's async transfers, then rendezvous with the other waves.
    asm volatile("s_wait_asynccnt 0" ::: "memory");
    __syncthreads();

    // ---- Sample tile from LDS: 4 lanes per output pixel (float4 of channels) ----
    const v4f* sm4 = (const v4f*)smem;
    for (int t = tid; t < TILE_H * TILE_W * 4; t += NTHREADS) {
        const int pix = t >> 2;
        const int cq  = t & 3;              // which float4 of the 16 channels
        const int py  = pix >> 5;           // / TILE_W
        const int px  = pix & (TILE_W - 1);
        const int ox  = ox0 + px;
        const int oy  = oy0 + py;

        const float x = mapx(ox);
        const float y = mapy(oy);

        // Reference semantics: trunc toward zero, x1 = x0+1 before clipping,
        // weights computed from clipped integer coords and unclipped floats.
        const int x0 = min(max((int)x, 0), 511);
        const int x1 = min(max((int)x + 1, 0), 511);
        const int y0 = min(max((int)y, 0), 511);
        const int y1 = min(max((int)y + 1, 0), 511);

        const float x0f = (float)x0, x1f = (float)x1;
        const float y0f = (float)y0, y1f = (float)y1;
        const float wa = (x1f - x) * (y1f - y);   // (y0,x0)
        const float wb = (x1f - x) * (y - y0f);   // (y1,x0)
        const float wc = (x - x0f) * (y1f - y);   // (y0,x1)
        const float wd = (x - x0f) * (y - y0f);   // (y1,x1)

        // Rect-local indices (defensive clamp; no-op for s < 1).
        const int ix0 = min(x0 - xlo, nCols - 1);
        const int ix1 = min(x1 - xlo, nCols - 1);
        const int iy0 = min(y0 - ylo, nRows - 1);
        const int iy1 = min(y1 - ylo, nRows - 1);

        const v4f pa = sm4[(iy0 * RECT_W + ix0) * 4 + cq];
        const v4f pb = sm4[(iy1 * RECT_W + ix0) * 4 + cq];
        const v4f pc = sm4[(iy0 * RECT_W + ix1) * 4 + cq];
        const v4f pd = sm4[(iy1 * RECT_W + ix1) * 4 + cq];

        v4f o;
        o.x = wa * pa.x + wb * pb.x + wc * pc.x + wd * pd.x;
        o.y = wa * pa.y + wb * pb.y + wc * pc.y + wd * pd.y;
        o.z = wa * pa.z + wb * pb.z + wc * pc.z + wd * pd.z;
        o.w = wa * pa.w + wb * pb.w + wc * pc.w + wd * pd.w;

        // Streaming output: non-temporal 128-bit store (don't pollute L2's
        // read-footprint residency).
        const size_t off = ((((size_t)b * 512 + (size_t)oy) * 512) + (size_t)ox) * 16
                           + (size_t)cq * 4;
        __builtin_nontemporal_store(o, (v4f*)(out + off));
    }
}

extern "C" void kernel_launch(void* const* d_in, const int* in_sizes, int n_in,
                              void* d_out, int out_size, void* d_ws, size_t ws_size,
                              hipStream_t stream) {
    const float* X         = (const float*)d_in[0];
    const float* scale     = (const float*)d_in[1];
    const float* translate = (const float*)d_in[2];
    float* out             = (float*)d_out;

    dim3 grid(512 / TILE_W, 512 / TILE_H, 16);   // (16, 32, 16) tiles
    bilinear_tile_kernel<<<grid, NTHREADS, 0, stream>>>(X, scale, translate, out);
}